// RoIAlign_17678085390505
// MI455X (gfx1250) — compile-verified
//
#include <hip/hip_runtime.h>

// RoIAlign for gfx1250 (MI455X), fp32, via V_WMMA_F32_16X16X4_F32 and the
// separable-bilinear factorization:
//   out[c,ph,pw] = sum_a sum_b Wy[ph,a] * Wx[pw,b] * feat[c, ry[a], cx[b]]
//
// Round-3 layout: WMMA K-dim = x-corner slots. The per-lane B operand pair
// (V0=K row 2hi, V1=K row 2hi+1) is exactly the two adjacent x-corner columns
// of one sample -> a single global_load_b64 yields the B operand in-layout
// (no register shuffles). A = masked block-diagonal Wx tiles (constant per
// wave); the y weight (validity + 1/S^2 folded) is a scalar pk_mul on the
// loaded pair. WMMA (y-slot a, K-group t) accumulates row pw=t of acc[ph=a/4]
// through the C operand; 196 WMMAs / 196 b64 gathers per wave.
//
// Memory-bound: the 164 MB feature map is L2-resident (192 MB L2); output is
// written once via nontemporal stores so it does not evict the feature map.

typedef __attribute__((ext_vector_type(2))) float v2f;
typedef __attribute__((ext_vector_type(8))) float v8f;
// 4-byte aligned pair: x-corner columns are adjacent but only dword-aligned.
typedef float v2fu __attribute__((ext_vector_type(2), aligned(4)));

#define POOLH  7
#define POOLW  7
#define HFEAT  200
#define WFEAT  200
#define CFEAT  256
#define SSCALE 0.25f
#define KY     28   // y: 7 bins * 2 samples * 2 corners
#define NXS    14   // x: 7 bins * 2 samples (adjacent corner pair per sample)

__global__ __launch_bounds__(256) void roialign_wmma_kernel(
    const float* __restrict__ feat,
    const float* __restrict__ rois,
    float* __restrict__ out)
{
    __shared__ int   s_xb  [NXS];  // clamped pair base column (<= W-2)
    __shared__ float s_wx0 [NXS];  // weight of column xb
    __shared__ float s_wx1 [NXS];  // weight of column xb+1
    __shared__ int   s_rowb[KY];   // ry * W * 4 : row byte offset per y slot
    __shared__ float s_wy  [KY];   // y weight (validity + 0.25 folded)
    __shared__ int   s_batch;

    const int roi = blockIdx.x;
    const int tid = threadIdx.x;

    // ---- per-RoI slot tables ----
    if (tid < KY) {
        const float* r = rois + (size_t)roi * 5;
        const float x1 = r[1] * SSCALE - 0.5f;
        const float y1 = r[2] * SSCALE - 0.5f;
        const float x2 = r[3] * SSCALE - 0.5f;
        const float y2 = r[4] * SSCALE - 0.5f;
        const float bw = (x2 - x1) * (1.0f / POOLW);
        const float bh = (y2 - y1) * (1.0f / POOLH);

        // y slots: tid = 2*sample + corner, sample = 2*ph + s
        {
            const int j      = tid;
            const int samp   = j >> 1;
            const int corner = j & 1;
            const int p      = samp >> 1;
            const int s      = samp & 1;
            const float fs   = ((float)s + 0.5f) * 0.5f;
            float y = y1 + ((float)p + fs) * bh;
            bool valid = (y > -1.0f) && (y < (float)HFEAT);
            float yc = fminf(fmaxf(y, 0.0f), (float)(HFEAT - 1));
            int y0 = (int)floorf(yc);
            int yi1 = y0 + 1; if (yi1 > HFEAT - 1) yi1 = HFEAT - 1;
            float ly = yc - (float)y0;
            float w  = corner ? ly : (1.0f - ly);
            if (!valid) w = 0.0f;
            s_rowb[j] = (corner ? yi1 : y0) * (WFEAT * 4);
            s_wy  [j] = w * 0.25f;           // fold 1/(S*S)
        }
        // x pair slots: tid = 2*pw + s  (14 threads)
        if (tid < NXS) {
            const int j  = tid;
            const int p  = j >> 1;
            const int s  = j & 1;
            const float fs = ((float)s + 0.5f) * 0.5f;
            float x = x1 + ((float)p + fs) * bw;
            bool valid = (x > -1.0f) && (x < (float)WFEAT);
            float xc = fminf(fmaxf(x, 0.0f), (float)(WFEAT - 1));
            int x0 = (int)floorf(xc);
            float lx = xc - (float)x0;
            float hx = 1.0f - lx;
            // pair base clamped so [xb, xb+1] stays in-row; when x0==W-1 the
            // high-corner weight is provably 0 (lx==0), so shift hx to slot 1.
            int   xb = x0;
            float w0 = hx, w1 = lx;
            if (x0 >= WFEAT - 1) { xb = WFEAT - 2; w0 = 0.0f; w1 = hx; }
            if (!valid) { w0 = 0.0f; w1 = 0.0f; }
            s_xb [j] = xb;
            s_wx0[j] = w0;
            s_wx1[j] = w1;
        }
        if (tid == 0) s_batch = (int)r[0];
    }
    __syncthreads();

    // ---- wave/channel-tile setup ----
    const int wave  = tid >> 5;
    const int lane  = tid & 31;
    const int nn    = lane & 15;     // WMMA M row (=pw in A/D), N col (=chan in B)
    const int hi    = lane >> 4;     // half-wave -> K rows {0,1} vs {2,3}
    const int ctile = blockIdx.y * 8 + wave;
    const int c     = ctile * 16 + nn;

    const int batch = __builtin_amdgcn_readfirstlane(s_batch);
    const float* __restrict__ fb =
        feat + (size_t)batch * (CFEAT * HFEAT * WFEAT);
    const unsigned chanoff = (unsigned)c * (HFEAT * WFEAT);

    // A tiles (constant): A_t[m,k] = Wx[m, 4t+k], nonzero only on row m == t.
    // Lane holds k = 2hi (V0) and 2hi+1 (V1) -> corners of x sample 2t+hi.
    // Per-lane base pointers: channel + x pair base of sample 2t+hi.
    v2f awx[7];
    const char* pt[7];
#pragma unroll
    for (int t = 0; t < 7; ++t) {
        const int xs = 2 * t + hi;
        awx[t].x = (nn == t) ? s_wx0[xs] : 0.0f;
        awx[t].y = (nn == t) ? s_wx1[xs] : 0.0f;
        pt[t] = (const char*)(fb + (chanoff + (unsigned)s_xb[xs]));
    }

    v8f acc[7];
#pragma unroll
    for (int p = 0; p < 7; ++p) acc[p] = (v8f){0.f,0.f,0.f,0.f,0.f,0.f,0.f,0.f};

    // ---- main contraction: 28 y slots, 7 x K-groups each ----
#pragma unroll
    for (int ph = 0; ph < 7; ++ph) {
#pragma unroll
        for (int aa = 0; aa < 4; ++aa) {
            const int      a    = 4 * ph + aa;
            const float    wya  = s_wy[a];
            const unsigned rowb = (unsigned)s_rowb[a];

            // 7 paired gathers for this y slot (issued as a clause)
            v2fu pr[7];
#pragma unroll
            for (int t = 0; t < 7; ++t)
                pr[t] = *(const v2fu*)(pt[t] + rowb);

            // B = wy[a] * pair (in-layout); chain into acc[ph]
#pragma unroll
            for (int t = 0; t < 7; ++t) {
                v2f bv; bv.x = wya * pr[t].x; bv.y = wya * pr[t].y;
                acc[ph] = __builtin_amdgcn_wmma_f32_16x16x4_f32(
                    false, awx[t], false, bv, (short)0, acc[ph], false, false);
            }
        }
    }

    // ---- store: D lane n (0-15) = channel c, element v = pw row ----
    if (lane < 16) {
        float* op = out + ((size_t)roi * CFEAT + (size_t)c) * (POOLH * POOLW);
#pragma unroll
        for (int ph = 0; ph < 7; ++ph) {
#pragma unroll
            for (int pw = 0; pw < 7; ++pw) {
                __builtin_nontemporal_store(acc[ph][pw], op + ph * POOLW + pw);
            }
        }
    }
}

extern "C" void kernel_launch(void* const* d_in, const int* in_sizes, int n_in,
                              void* d_out, int out_size, void* d_ws, size_t ws_size,
                              hipStream_t stream) {
    const float* feat = (const float*)d_in[0];
    const float* rois = (const float*)d_in[1];
    float* out = (float*)d_out;

    const int K = in_sizes[1] / 5;         // number of RoIs
    // 256 channels = grid.y(2) * 8 waves * 16 channels/wave
    dim3 grid((unsigned)K, 2, 1);
    dim3 block(256, 1, 1);
    roialign_wmma_kernel<<<grid, block, 0, stream>>>(feat, rois, out);
}